// HomogeneousPoissonEncoder_26749056319601
// MI455X (gfx1250) — compile-verified
//
#include <hip/hip_runtime.h>

// Homogeneous Poisson encoder for MI455X (gfx1250).
// Memory-bound: ~256MB output write + ~35MB effective noise read (early exit
// once cumulative spike time passes the 500ms horizon). Target ~12.5us.
//
// CDNA5 paths used: global_load_async_to_lds_b128 + s_wait_asynccnt double
// buffering for the latency-sensitive sequential interval scan, ds_or_b32
// bitmask scratch in LDS (wave32 conflict-free layout), non-temporal b128
// stores for the streamed output.

typedef float v4f __attribute__((ext_vector_type(4)));
typedef int   v4i __attribute__((ext_vector_type(4)));

#define STEPS   500
#define BN      (32 * 4096)          // batch * neurons
#define ELEMS   4                    // columns per thread
#define BLOCK   128                  // threads per block (4 waves)
#define NWORDS  16                   // 512 bits of spike mask per element
#define GSTEPS  4                    // pipeline group size (steps)
#define NGROUPS (STEPS / GSTEPS)     // 125

#if defined(__HIP_DEVICE_COMPILE__) && \
    __has_builtin(__builtin_amdgcn_global_load_async_to_lds_b128)
#define USE_ASYNC 1
#else
#define USE_ASYNC 0
#endif

#if USE_ASYNC
#if __has_builtin(__builtin_amdgcn_s_wait_asynccnt)
#define WAIT_ASYNC(n) __builtin_amdgcn_s_wait_asynccnt(n)
#else
#define WAIT_ASYNC(n) asm volatile("s_wait_asynccnt %0" ::"i"(n) : "memory")
#endif
// Typed, address-space-qualified pointers matching the builtin signature:
//   void (v4i addrspace(1)* src, v4i addrspace(3)* dst, imm off, imm cpol)
typedef __attribute__((address_space(1))) v4i gas_v4i;
typedef __attribute__((address_space(3))) v4i las_v4i;
#endif

__global__ __launch_bounds__(BLOCK)
void poisson_encode_kernel(const float* __restrict__ inputs,
                           const float* __restrict__ noise,
                           float* __restrict__ out)
{
    // 32 KB spike bitmask: word W of thread tid at smask[W*BLOCK + tid]
    // -> wave lanes always hit distinct LDS banks (bank = tid % 64).
    __shared__ unsigned smask[NWORDS * ELEMS * BLOCK];
#if USE_ASYNC
    __shared__ v4f snoise[2][GSTEPS * BLOCK];   // 2 x 8 KB staging buffers
#endif

    const int tid  = threadIdx.x;
    const int col0 = (blockIdx.x * BLOCK + tid) * ELEMS;

    // Clear this thread's bitmask words (private -> no barrier needed).
    #pragma unroll
    for (int w = 0; w < NWORDS * ELEMS; ++w)
        smask[w * BLOCK + tid] = 0u;

    v4f u = *(const v4f*)(inputs + col0);
    float mn[ELEMS], t[ELEMS];
    #pragma unroll
    for (int e = 0; e < ELEMS; ++e) {
        float ue  = u[e];
        bool  act = ue > 0.0f;
        // mean_ms = 1000/(100*u) - refrac = 10/u - 1  (compensated)
        mn[e] = act ? (10.0f / ue - 1.0f) : 0.0f;
        t[e]  = act ? 0.0f : 1.0e9f;    // inactive -> overflow bin forever
    }

    // One inter-spike interval per element: t += noise*mean + refrac,
    // bin = ceil(t/dt) - 1, record bin if inside horizon.
    auto step = [&](v4f nz) {
        #pragma unroll
        for (int e = 0; e < ELEMS; ++e) {
            t[e] = fmaf(nz[e], mn[e], t[e] + 1.0f);
            float tc = fminf(t[e], 1024.0f);      // keep int conversion safe
            int idx = (int)ceilf(tc) - 1;
            if (idx < STEPS)
                atomicOr(&smask[(((unsigned)idx >> 5) * ELEMS + (unsigned)e)
                                 * BLOCK + tid],
                         1u << (idx & 31));        // ds_or_b32, no return
        }
    };

#if USE_ASYNC
    // Double-buffered async global->LDS prefetch (ASYNCcnt pipeline).
    auto issue_group = [&](int g, int b) {
        #pragma unroll
        for (int j = 0; j < GSTEPS; ++j) {
            const float* src = noise + (size_t)(g * GSTEPS + j) * BN + col0;
            v4i* gsrc = (v4i*)src;                       // drop const, retype
            v4i* ldst = (v4i*)(&snoise[b][j * BLOCK + tid]);
            __builtin_amdgcn_global_load_async_to_lds_b128(
                (gas_v4i*)gsrc, (las_v4i*)ldst, 0, 0);
        }
    };
    issue_group(0, 0);
    for (int g = 0; g < NGROUPS; ++g) {
        if (g + 1 < NGROUPS) {
            issue_group(g + 1, (g + 1) & 1);
            WAIT_ASYNC(GSTEPS);   // group g's 4 async ops have landed
        } else {
            WAIT_ASYNC(0);
        }
        #pragma unroll
        for (int j = 0; j < GSTEPS; ++j)
            step(snoise[g & 1][j * BLOCK + tid]);
        // All 4 chains past the 500ms horizon -> later spikes all overflow.
        if (t[0] >= (float)STEPS && t[1] >= (float)STEPS &&
            t[2] >= (float)STEPS && t[3] >= (float)STEPS)
            break;
    }
#else
    // Fallback: direct unrolled non-temporal b128 loads.
    for (int s = 0; s < STEPS; s += GSTEPS) {
        v4f nz[GSTEPS];
        #pragma unroll
        for (int j = 0; j < GSTEPS; ++j)
            nz[j] = __builtin_nontemporal_load(
                (const v4f*)(noise + (size_t)(s + j) * BN + col0));
        #pragma unroll
        for (int j = 0; j < GSTEPS; ++j)
            step(nz[j]);
        if (t[0] >= (float)STEPS && t[1] >= (float)STEPS &&
            t[2] >= (float)STEPS && t[3] >= (float)STEPS)
            break;
    }
#endif

    // Phase 2: expand bitmask to 0/1 floats; coalesced NT b128 stores
    // (512 B per wave per step). Covers every output element, so no
    // separate zero-fill pass is needed.
    for (int w = 0; w < NWORDS; ++w) {
        unsigned m[ELEMS];
        #pragma unroll
        for (int e = 0; e < ELEMS; ++e)
            m[e] = smask[(w * ELEMS + e) * BLOCK + tid];
        int nb = STEPS - w * 32;
        if (nb > 32) nb = 32;
        #pragma unroll 8
        for (int j = 0; j < nb; ++j) {
            v4f v;
            #pragma unroll
            for (int e = 0; e < ELEMS; ++e)
                v[e] = ((m[e] >> j) & 1u) ? 1.0f : 0.0f;
            __builtin_nontemporal_store(
                v, (v4f*)(out + (size_t)(w * 32 + j) * BN + col0));
        }
    }
}

extern "C" void kernel_launch(void* const* d_in, const int* in_sizes, int n_in,
                              void* d_out, int out_size, void* d_ws,
                              size_t ws_size, hipStream_t stream)
{
    (void)in_sizes; (void)n_in; (void)out_size; (void)d_ws; (void)ws_size;
    const float* inputs = (const float*)d_in[0];   // [32, 4096]
    const float* noise  = (const float*)d_in[1];   // [500, 32, 4096]
    float* out = (float*)d_out;                    // [500, 32, 4096]

    const int grid = BN / (BLOCK * ELEMS);         // 256 blocks
    poisson_encode_kernel<<<grid, BLOCK, 0, stream>>>(inputs, noise, out);
}